// RegisterRotation2D_67740224192655
// MI455X (gfx1250) — compile-verified
//
#include <hip/hip_runtime.h>
#include <math.h>

typedef __attribute__((ext_vector_type(2))) float v2f;
typedef __attribute__((ext_vector_type(8))) float v8f;

#define HH 160
#define WW 160
#define NPIX (HH * WW)
#define NB 8
#define CT 23        // coarse angle tiles of 16 -> 368 slots (>= 360)
#define SLAB_ROWS 8  // rows per slab
#define NS 20        // slabs (160/8) -> 8*23*20 = 3680 coarse waves
#define FT 2         // fine angle tiles of 16 -> 32 slots (>= 21)

// One bilinear sample of img at the inverse-rotated coordinate of (y,x).
// cdy = cy + c*dy, sdy = cx + s*dy are row-invariant; dx varies per pixel.
__device__ __forceinline__ float bilin_sample(const float* __restrict__ img,
                                              float c, float s,
                                              float cdy, float sdy, float dx) {
  float sy = fmaf(-s, dx, cdy);
  float sx = fmaf( c, dx, sdy);
  float fy = floorf(sy), fx = floorf(sx);
  float wy = sy - fy,   wx = sx - fx;
  int iy0 = (int)fy; iy0 = iy0 < 0 ? 0 : (iy0 > HH - 1 ? HH - 1 : iy0);
  int iy1 = iy0 + 1 > HH - 1 ? HH - 1 : iy0 + 1;
  int ix0 = (int)fx; ix0 = ix0 < 0 ? 0 : (ix0 > WW - 1 ? WW - 1 : ix0);
  int ix1 = ix0 + 1 > WW - 1 ? WW - 1 : ix0 + 1;
  const float* r0 = img + iy0 * WW;
  const float* r1 = img + iy1 * WW;
  float v00 = r0[ix0], v01 = r0[ix1], v10 = r1[ix0], v11 = r1[ix1];
  float top = fmaf(wx, v01 - v00, v00);
  float bot = fmaf(wx, v11 - v10, v10);
  float val = fmaf(wy, bot - top, top);
  bool valid = (sy >= 0.0f) && (sy <= (float)(HH - 1)) &&
               (sx >= 0.0f) && (sx <= (float)(WW - 1));
  return valid ? val : 0.0f;
}

// One wave32 computes partial correlations for 16 angles (M dim of WMMA)
// over an 8-row slab, accumulating via V_WMMA_F32_16X16X4_F32 (K = 4 pixels
// per step, full f32 precision). theta_deg is per-lane (angle of lane&15).
__device__ __forceinline__ void corr_tile_wave(const float* __restrict__ src,
                                               const float* __restrict__ tgt,
                                               float theta_deg, int yStart,
                                               float* __restrict__ outp) {
  const int lane = threadIdx.x;
  const int k0 = (lane >> 4) << 1;          // lanes 0-15: K=0,1 ; 16-31: K=2,3
  const float th = theta_deg * 0.017453292519943295f;
  const float c = cosf(th), s = sinf(th);
  const float cy = 79.5f;

  v8f acc = {};
  for (int y = yStart; y < yStart + SLAB_ROWS; ++y) {
    float dy  = (float)y - cy;
    float cdy = fmaf(c, dy, cy);            // cy + c*dy
    float sdy = fmaf(s, dy, cy);            // cx + s*dy  (cx == cy)
    const float* trow = tgt + y * WW;
    for (int xb = 0; xb < WW; xb += 4) {
      int x0 = xb + k0;
      v2f a, b;
      a.x = bilin_sample(src, c, s, cdy, sdy, (float)x0 - cy);
      a.y = bilin_sample(src, c, s, cdy, sdy, (float)(x0 + 1) - cy);
      b.x = trow[x0];                        // B[k][n] = target[pixel k], all n
      b.y = trow[x0 + 1];
      acc = __builtin_amdgcn_wmma_f32_16x16x4_f32(
          false, a, false, b, (short)0, acc, false, false);
    }
  }
  // D layout: VGPR r holds (M=r, lanes 0-15) / (M=r+8, lanes 16-31);
  // all 16 N columns are identical, so lanes 0 and 16 carry everything.
  if (lane == 0) {
#pragma unroll
    for (int i = 0; i < 8; ++i) outp[i] = acc[i];
  } else if (lane == 16) {
#pragma unroll
    for (int i = 0; i < 8; ++i) outp[8 + i] = acc[i];
  }
}

__global__ void __launch_bounds__(32)
coarse_partial(const float* __restrict__ src, const float* __restrict__ tgt,
               float* __restrict__ partial) {
  int b = blockIdx.x, tile = blockIdx.y, slab = blockIdx.z;
  int m = threadIdx.x & 15;
  __builtin_prefetch(src + b * NPIX, 0, 3);
  __builtin_prefetch(tgt + b * NPIX, 0, 3);
  float theta = (float)(tile * 16 + m);     // ROTATION_RESOLUTION=1, GUESS=0
  corr_tile_wave(src + b * NPIX, tgt + b * NPIX, theta, slab * SLAB_ROWS,
                 partial + ((((b * CT + tile) * NS) + slab) << 4));
}

__global__ void __launch_bounds__(128)
coarse_argmax(const float* __restrict__ partial, float* __restrict__ baseAng) {
  __shared__ float sv[128];
  __shared__ int   si[128];
  int b = blockIdx.x, t = threadIdx.x;
  float bv = -INFINITY; int bi = 0x7fffffff;
  for (int a = t; a < 360; a += 128) {      // slots 360..367 excluded
    int tile = a >> 4, m = a & 15;
    const float* p = partial + (((b * CT + tile) * NS) << 4) + m;
    float sum = 0.0f;
    for (int sl = 0; sl < NS; ++sl) sum += p[sl * 16];   // fixed order
    if (sum > bv || (sum == bv && a < bi)) { bv = sum; bi = a; }
  }
  sv[t] = bv; si[t] = bi; __syncthreads();
  for (int off = 64; off > 0; off >>= 1) {
    if (t < off) {
      if (sv[t + off] > sv[t] || (sv[t + off] == sv[t] && si[t + off] < si[t])) {
        sv[t] = sv[t + off]; si[t] = si[t + off];
      }
    }
    __syncthreads();
  }
  if (t == 0) baseAng[b] = (float)si[0];    // *1.0 deg
}

__global__ void __launch_bounds__(32)
fine_partial(const float* __restrict__ src, const float* __restrict__ tgt,
             const float* __restrict__ baseAng, float* __restrict__ partial) {
  int b = blockIdx.x, tile = blockIdx.y, slab = blockIdx.z;
  int m = threadIdx.x & 15;
  int fi = tile * 16 + m; if (fi > 20) fi = 20;   // pad slots dup index 20
  float theta = baseAng[b] + (float)(fi - 10) * 0.1f;
  corr_tile_wave(src + b * NPIX, tgt + b * NPIX, theta, slab * SLAB_ROWS,
                 partial + ((((b * FT + tile) * NS) + slab) << 4));
}

__global__ void __launch_bounds__(32)
fine_argmax(const float* __restrict__ partial, const float* __restrict__ baseAng,
            float* __restrict__ out) {
  __shared__ float sv[32];
  __shared__ int   si[32];
  int b = blockIdx.x, t = threadIdx.x;
  float v = -INFINITY;
  if (t < 21) {
    int tile = t >> 4, m = t & 15;
    const float* p = partial + (((b * FT + tile) * NS) << 4) + m;
    v = 0.0f;
    for (int sl = 0; sl < NS; ++sl) v += p[sl * 16];
  }
  sv[t] = v; si[t] = t; __syncthreads();
  for (int off = 16; off > 0; off >>= 1) {
    if (t < off) {
      if (sv[t + off] > sv[t] || (sv[t + off] == sv[t] && si[t + off] < si[t])) {
        sv[t] = sv[t + off]; si[t] = si[t + off];
      }
    }
    __syncthreads();
  }
  if (t == 0) {
    float ang = baseAng[b] + (float)(si[0] - 10) * 0.1f;
    float w = fmodf(ang + 180.0f, 360.0f);
    if (w < 0.0f) w += 360.0f;
    out[b] = w - 180.0f;
  }
}

extern "C" void kernel_launch(void* const* d_in, const int* in_sizes, int n_in,
                              void* d_out, int out_size, void* d_ws, size_t ws_size,
                              hipStream_t stream) {
  (void)in_sizes; (void)n_in; (void)out_size; (void)ws_size;
  const float* target = (const float*)d_in[0];   // setup_inputs order: target, src
  const float* src    = (const float*)d_in[1];
  float* out = (float*)d_out;

  float* pCoarse = (float*)d_ws;                         // 8*23*20*16 = 58880
  float* baseAng = pCoarse + NB * CT * NS * 16;          // 8
  float* pFine   = baseAng + NB;                         // 8*2*20*16  = 5120

  coarse_partial<<<dim3(NB, CT, NS), 32, 0, stream>>>(src, target, pCoarse);
  coarse_argmax <<<NB, 128, 0, stream>>>(pCoarse, baseAng);
  fine_partial  <<<dim3(NB, FT, NS), 32, 0, stream>>>(src, target, baseAng, pFine);
  fine_argmax   <<<NB, 32, 0, stream>>>(pFine, baseAng, out);
}